// MembershipDecoder_41403484733992
// MI455X (gfx1250) — compile-verified
//
#include <hip/hip_runtime.h>
#include <math.h>

typedef __attribute__((ext_vector_type(2))) float v2f;
typedef __attribute__((ext_vector_type(8))) float v8f;

#define N_ROWS 1024
#define D_IN   128
#define H      100
#define H_PAD  112
#define LN_EPS 1e-3f

// ---------------------------------------------------------------------------
// Kernel 0: fold LN-affine + W_m + W_o into a single weight vector.
//   w[k]  = dot(W_m[k,:], W_o)           g[k] = gamma[k]*w[k]
//   consts[0] = sum(g)                   consts[1] = dot(beta,w) + dot(b_m,W_o) + b_o
// ---------------------------------------------------------------------------
__global__ void precompute_kernel(const float* __restrict__ gamma,
                                  const float* __restrict__ beta,
                                  const float* __restrict__ W_m,
                                  const float* __restrict__ b_m,
                                  const float* __restrict__ W_o,
                                  const float* __restrict__ b_o,
                                  float* __restrict__ g,
                                  float* __restrict__ consts) {
    __shared__ float sg[128];
    __shared__ float sc[128];
    const int t = threadIdx.x;  // 0..127
    float gk = 0.f, ck = 0.f;
    if (t < H) {
        float wk = 0.f;
        for (int j = 0; j < H; ++j) wk = fmaf(W_m[t * H + j], W_o[j], wk);
        gk = gamma[t] * wk;
        g[t] = gk;
        ck = beta[t] * wk + b_m[t] * W_o[t];
    }
    sg[t] = gk; sc[t] = ck;
    __syncthreads();
    for (int s = 64; s > 0; s >>= 1) {
        if (t < s) { sg[t] += sg[t + s]; sc[t] += sc[t + s]; }
        __syncthreads();
    }
    if (t == 0) { consts[0] = sg[0]; consts[1] = sc[0] + b_o[0]; }
}

// ---------------------------------------------------------------------------
// Kernel 0b: transpose + zero-pad the GEMM weights.
//   Wt[n][k] = W[k][n] for n < 100, else 0.   Wt is [112 x 128].
// This makes the WMMA B-fragment a single contiguous b64 load per K-step
// with no predication inside the hot loop.
// grid = (2 matrices, 7 chunks), block = 256.
// ---------------------------------------------------------------------------
__global__ void pad_weights_kernel(const float* __restrict__ W_s,
                                   const float* __restrict__ W_c,
                                   float* __restrict__ Wt_s,
                                   float* __restrict__ Wt_c) {
    const bool isC = (blockIdx.x != 0);
    const float* __restrict__ W = isC ? W_c : W_s;
    float* __restrict__ Wt      = isC ? Wt_c : Wt_s;
    const int base = blockIdx.y * 2048;          // 7 * 2048 == 112*128
    for (int i = base + threadIdx.x; i < base + 2048; i += 256) {
        const int n = i / D_IN;
        const int k = i % D_IN;
        Wt[i] = (n < H) ? W[k * H + n] : 0.f;
    }
}

// ---------------------------------------------------------------------------
// Kernel 1: WMMA f32 GEMM.  out[1024 x 100] = X[1024 x 128] @ W[128 x 100] + b
// One wave32 per 16x16 output tile; K-loop of 32 x V_WMMA_F32_16X16X4_F32,
// each step = two unconditional global_load_b64 + one WMMA.
// grid = (64 row-tiles, 7 col-tiles, 2 matrices), block = 32.
// A layout (16x4 f32): lanes 0-15 rows M=0..15 K={0,1}; lanes 16-31 K={2,3}.
// B layout (4x16 f32): mirror with N in place of M (fed from transposed Wt).
// D layout (16x16 f32): vgpr r -> M = r + (lane>=16 ? 8:0), N = lane&15.
// ---------------------------------------------------------------------------
__global__ void gemm_wmma_kernel(const float* __restrict__ seq,
                                 const float* __restrict__ col,
                                 const float* __restrict__ Wt_s,
                                 const float* __restrict__ b_s,
                                 const float* __restrict__ Wt_c,
                                 const float* __restrict__ b_c,
                                 float* __restrict__ s_buf,
                                 float* __restrict__ c_buf) {
    const int lane = threadIdx.x;                // 0..31
    const int mt = blockIdx.x;                   // 0..63
    const int nt = blockIdx.y;                   // 0..6
    const bool isC = (blockIdx.z != 0);
    const float* __restrict__ X    = isC ? col : seq;
    const float* __restrict__ Wt   = isC ? Wt_c : Wt_s;
    const float* __restrict__ bias = isC ? b_c : b_s;
    float* __restrict__ out        = isC ? c_buf : s_buf;

    const int idx16 = lane & 15;
    const int koff  = (lane >> 4) * 2;           // 0 or 2
    const float* __restrict__ arow = X  + (mt * 16 + idx16) * D_IN + koff;
    const float* __restrict__ brow = Wt + (nt * 16 + idx16) * D_IN + koff;

    v8f acc = {};
    for (int k0 = 0; k0 < D_IN; k0 += 4) {
        const v2f a = *(const v2f*)(arow + k0);  // 8B-aligned (koff even)
        const v2f b = *(const v2f*)(brow + k0);
        acc = __builtin_amdgcn_wmma_f32_16x16x4_f32(
            /*neg_a=*/false, a, /*neg_b=*/false, b,
            /*c_mod=*/(short)0, acc, /*reuse_a=*/false, /*reuse_b=*/false);
    }

    const int dcol  = nt * 16 + idx16;
    const int rbase = (lane >> 4) * 8;
    if (dcol < H) {
        const float bv = bias[dcol];
#pragma unroll
        for (int r = 0; r < 8; ++r)
            out[(mt * 16 + rbase + r) * H + dcol] = acc[r] + bv;
    }
}

// ---------------------------------------------------------------------------
// Kernel 2: pairwise fused relu + LayerNorm + folded-linear -> logits.
// One block = 16x16 (n,m) pairs; s/c tiles + g staged in LDS.
// stride-100 LDS rows give a conflict-free bank pattern (100 mod 64 = 36,
// 36*tx mod 64 distinct for tx in 0..15).
// grid = (64 m-tiles, 64 n-tiles), block = 256.
// ---------------------------------------------------------------------------
__global__ void logits_kernel(const float* __restrict__ s_buf,
                              const float* __restrict__ c_buf,
                              const float* __restrict__ g,
                              const float* __restrict__ consts,
                              float* __restrict__ logits) {
    __shared__ float s_sm[16 * H];
    __shared__ float c_sm[16 * H];
    __shared__ float g_sm[H];
    const int t  = threadIdx.x;                  // 0..255
    const int mt = blockIdx.x;                   // m tile
    const int nt = blockIdx.y;                   // n tile

    for (int i = t; i < 16 * H; i += 256) {
        const int r = i / H, k = i % H;
        s_sm[i] = s_buf[(nt * 16 + r) * H + k];
        c_sm[i] = c_buf[(mt * 16 + r) * H + k];
    }
    if (t < H) g_sm[t] = g[t];
    __syncthreads();

    const float sumg = consts[0];
    const float Cc   = consts[1];
    const int ml = t & 15;
    const int nl = t >> 4;
    const float* __restrict__ srow = &s_sm[nl * H];
    const float* __restrict__ crow = &c_sm[ml * H];

    float S1 = 0.f, S2 = 0.f, Sg = 0.f;
#pragma unroll 4
    for (int k = 0; k < H; ++k) {
        float r = srow[k] + crow[k];
        r = fmaxf(r, 0.f);
        S1 += r;
        S2 = fmaf(r, r, S2);
        Sg = fmaf(r, g_sm[k], Sg);
    }
    const float mu  = S1 * (1.0f / H);
    const float var = S2 * (1.0f / H) - mu * mu;
    const float inv = rsqrtf(var + LN_EPS);
    const float logit = inv * (Sg - mu * sumg) + Cc;
    logits[(nt * 16 + nl) * N_ROWS + mt * 16 + ml] = logit;
}

// ---------------------------------------------------------------------------
// Kernel 3: row softmax, in place over d_out[0:N*N), duplicated to 2nd half.
// grid = 1024 rows, block = 256 (4 elements/thread).
// ---------------------------------------------------------------------------
__global__ void softmax_kernel(float* __restrict__ out) {
    __shared__ float red[256];
    const int row = blockIdx.x;
    const int t = threadIdx.x;
    float* __restrict__ lr = out + (size_t)row * N_ROWS;
    float v[4];
    float mx = -INFINITY;
#pragma unroll
    for (int i = 0; i < 4; ++i) { v[i] = lr[t + i * 256]; mx = fmaxf(mx, v[i]); }
    red[t] = mx; __syncthreads();
    for (int s = 128; s > 0; s >>= 1) {
        if (t < s) red[t] = fmaxf(red[t], red[t + s]);
        __syncthreads();
    }
    mx = red[0]; __syncthreads();
    float sum = 0.f;
#pragma unroll
    for (int i = 0; i < 4; ++i) { v[i] = __expf(v[i] - mx); sum += v[i]; }
    red[t] = sum; __syncthreads();
    for (int s = 128; s > 0; s >>= 1) {
        if (t < s) red[t] += red[t + s];
        __syncthreads();
    }
    const float rinv = 1.0f / red[0];
    float* __restrict__ out2 = out + (size_t)N_ROWS * N_ROWS + (size_t)row * N_ROWS;
#pragma unroll
    for (int i = 0; i < 4; ++i) {
        const float p = v[i] * rinv;
        lr[t + i * 256]   = p;
        out2[t + i * 256] = p;
    }
}

// ---------------------------------------------------------------------------
// Input order: 0 seq, 1 col, 2 seq_lens(unused), 3 W_s, 4 b_s, 5 W_c, 6 b_c,
//              7 gamma, 8 beta, 9 W_m, 10 b_m, 11 W_o, 12 b_o
// ---------------------------------------------------------------------------
extern "C" void kernel_launch(void* const* d_in, const int* in_sizes, int n_in,
                              void* d_out, int out_size, void* d_ws, size_t ws_size,
                              hipStream_t stream) {
    const float* seq   = (const float*)d_in[0];
    const float* col   = (const float*)d_in[1];
    const float* W_s   = (const float*)d_in[3];
    const float* b_s   = (const float*)d_in[4];
    const float* W_c   = (const float*)d_in[5];
    const float* b_c   = (const float*)d_in[6];
    const float* gamma = (const float*)d_in[7];
    const float* beta  = (const float*)d_in[8];
    const float* W_m   = (const float*)d_in[9];
    const float* b_m   = (const float*)d_in[10];
    const float* W_o   = (const float*)d_in[11];
    const float* b_o   = (const float*)d_in[12];
    float* out = (float*)d_out;

    float* s_buf  = (float*)d_ws;                 // [1024*100]
    float* c_buf  = s_buf + N_ROWS * H;           // [1024*100]
    float* g      = c_buf + N_ROWS * H;           // [128]
    float* consts = g + 128;                      // [16]
    float* Wt_s   = consts + 16;                  // [112*128]
    float* Wt_c   = Wt_s + H_PAD * D_IN;          // [112*128]

    precompute_kernel<<<1, 128, 0, stream>>>(gamma, beta, W_m, b_m, W_o, b_o, g, consts);

    dim3 pgrid(2, 7);
    pad_weights_kernel<<<pgrid, 256, 0, stream>>>(W_s, W_c, Wt_s, Wt_c);

    dim3 ggrid(N_ROWS / 16, H_PAD / 16, 2);       // 64 x 7 x 2
    gemm_wmma_kernel<<<ggrid, 32, 0, stream>>>(seq, col, Wt_s, b_s, Wt_c, b_c, s_buf, c_buf);

    dim3 lgrid(N_ROWS / 16, N_ROWS / 16);         // 64 x 64
    logits_kernel<<<lgrid, 256, 0, stream>>>(s_buf, c_buf, g, consts, out);

    softmax_kernel<<<N_ROWS, 256, 0, stream>>>(out);
}